// PureCascadedBitFFN_62380105007549
// MI455X (gfx1250) — compile-verified
//
#include <hip/hip_runtime.h>

// PureCascadedBitFFN — gfx1250 (MI455X) streaming bit-extraction kernel.
//
// Reference math: 16-step MSB->LSB threshold cascade with hard sigmoid gate.
// Exact equivalence (proved from fp32 exactness of the cascade on inputs that
// are multiples of 2^-8 below 2^16):   n = ceil(x - 0.5),  out[j] = bit j of n.
//
// Memory-bound: 8 MB in + 128 MB out = 136 MB  ->  ~5.8 us at 23.3 TB/s.
// Strategy: one lane per input element; pack the 16 bits into an int, then use
// wave32 ds_bpermute to redistribute so each global_store_b128 is a fully
// contiguous 512 B per wave. Non-temporal hints on the streamed input/output.

typedef float v4f __attribute__((ext_vector_type(4)));

__global__ __launch_bounds__(256) void
PureCascadedBitFFN_62380105007549_kernel(const float* __restrict__ in,
                                         float* __restrict__ out,
                                         int n_elems) {
    const int t    = blockIdx.x * blockDim.x + threadIdx.x;
    const int lane = t & 31;          // wave32: lanes 0..31
    const int eb   = t & ~31;         // first element owned by this wave

    if (eb + 32 <= n_elems) {
        // ---- fast path: full wave, fully coalesced ----
        // Coalesced 128 B NT load: lane L reads element eb+L (single use).
        float x = __builtin_nontemporal_load(in + eb + lane);

        // n = ceil(x - 0.5): exact integer equivalent of the hard cascade.
        int n = (int)__builtin_ceilf(x - 0.5f);

        const int j0 = (lane & 3) * 4;                 // this lane's 4 bit slots
        float* ow = out + (size_t)eb * 16;             // wave's 2 KB output chunk

        #pragma unroll
        for (int k = 0; k < 4; ++k) {
            // Lane L needs the packed bits of element eb + 8k + (L>>2).
            int srcLane = 8 * k + (lane >> 2);
            int m = __builtin_amdgcn_ds_bpermute(srcLane << 2, n);  // wave32 lane gather

            v4f v;
            v.x = (float)((m >> (j0 + 0)) & 1);
            v.y = (float)((m >> (j0 + 1)) & 1);
            v.z = (float)((m >> (j0 + 2)) & 1);
            v.w = (float)((m >> (j0 + 3)) & 1);

            // 512 B contiguous per wave per instruction; NT: pure streaming.
            __builtin_nontemporal_store(v, (v4f*)(ow + k * 128 + lane * 4));
        }
    } else {
        // ---- tail path (not hit for 512*4096, kept for safety) ----
        int e = eb + lane;
        if (e < n_elems) {
            float x = in[e];
            int n = (int)__builtin_ceilf(x - 0.5f);
            float* o = out + (size_t)e * 16;
            #pragma unroll
            for (int j = 0; j < 16; ++j)
                o[j] = (float)((n >> j) & 1);
        }
    }
}

extern "C" void kernel_launch(void* const* d_in, const int* in_sizes, int n_in,
                              void* d_out, int out_size, void* d_ws, size_t ws_size,
                              hipStream_t stream) {
    (void)n_in; (void)out_size; (void)d_ws; (void)ws_size;
    const float* in  = (const float*)d_in[0];
    float*       out = (float*)d_out;
    const int n = in_sizes[0];                 // 512*4096 = 2,097,152

    const int threads = 256;                   // 8 wave32s per block
    const int blocks  = (n + threads - 1) / threads;
    PureCascadedBitFFN_62380105007549_kernel<<<blocks, threads, 0, stream>>>(in, out, n);
}